// MultiHeadAttentionLayer_38989713113820
// MI455X (gfx1250) — compile-verified
//
#include <hip/hip_runtime.h>
#include <hip/hip_bf16.h>
#include <math.h>

// Problem constants (match reference): N=50000, E=1600000, IN_DIM=128, H=8, D=16
#define IN_DIM  128
#define OUT_DIM 128   // H*D
#define NHEAD   8
#define HDIM    16
#define LDS_PITCH 132  // 128 + 4 pad floats: lane l -> bank 4l mod 64, conflict-free

typedef __attribute__((ext_vector_type(2))) float v2f;
typedef __attribute__((ext_vector_type(8))) float v8f;
typedef int v4i_vec __attribute__((vector_size(4 * sizeof(int))));

// ---------------------------------------------------------------------------
// Zero-init scratch accumulators (wV and z are contiguous in workspace).
// ---------------------------------------------------------------------------
__global__ void zero_f32(float* __restrict__ p, long long n) {
  long long i = (long long)blockIdx.x * blockDim.x + threadIdx.x;
  if (i < n) p[i] = 0.0f;
}

// ---------------------------------------------------------------------------
// Fused QKV projection GEMM using V_WMMA_F32_16X16X4_F32 (exact fp32 math).
//   blockDim = 256 (8 waves). The block's 16x128 A-tile of h is staged once
//   into LDS via the gfx1250 async global->LDS path (ASYNCcnt), then each
//   wave computes one 16x16 output tile; blockIdx.y selects Q/K/V.
// fp32 WMMA VGPR layout (ISA 7.12.2):
//   A(16x4):  lane&15 = M, lane>>4 picks K-pair; a = {A[M][k0+2h], A[M][k0+2h+1]}
//   B(4x16):  lane&15 = N, lane>>4 picks K-pair (symmetric to A)
//   C/D(16x16): VGPR v holds row v + 8*(lane>>4), col lane&15
// ---------------------------------------------------------------------------
__global__ __launch_bounds__(256) void qkv_gemm_wmma(
    const float* __restrict__ h,
    const float* __restrict__ Wq, const float* __restrict__ bq,
    const float* __restrict__ Wk, const float* __restrict__ bk,
    const float* __restrict__ Wv, const float* __restrict__ bv,
    float* __restrict__ Qb, float* __restrict__ Kb, float* __restrict__ Vb,
    int n_nodes)
{
  __shared__ float tileA[16 * LDS_PITCH];   // 8.25 KB of the 320 KB WGP LDS

  const float* W;  const float* bias;  float* out;
  if (blockIdx.y == 0)      { W = Wq; bias = bq; out = Qb; }
  else if (blockIdx.y == 1) { W = Wk; bias = bk; out = Kb; }
  else                      { W = Wv; bias = bv; out = Vb; }

  const int tid  = threadIdx.x;
  const int wave = tid >> 5;
  const int lane = tid & 31;
  const int half = lane >> 4;      // 0: K pair {0,1}, 1: K pair {2,3}
  const int l15  = lane & 15;

  const int row0 = blockIdx.x * 16;   // N = 50000 = 3125*16, exact
  const int n0   = wave * 16;         // 8 waves x 16 = 128 output cols
  if (row0 >= n_nodes) return;

  // ---- Stage A tile: 16 rows x 128 floats = 512 float4, 2 per thread ----
  {
    const float* gsrc = h + (size_t)row0 * IN_DIM;
#if __has_builtin(__builtin_amdgcn_global_load_async_to_lds_b128)
    #pragma unroll
    for (int i = tid; i < 512; i += 256) {
      const int r = i >> 5;        // row 0..15
      const int c = i & 31;        // float4 index within row
      __builtin_amdgcn_global_load_async_to_lds_b128(
          (__attribute__((address_space(1))) v4i_vec*)(gsrc + (size_t)r * IN_DIM + c * 4),
          (__attribute__((address_space(3))) v4i_vec*)(tileA + r * LDS_PITCH + c * 4),
          0, 0);
    }
#if __has_builtin(__builtin_amdgcn_s_wait_asynccnt)
    __builtin_amdgcn_s_wait_asynccnt(0);
#else
    asm volatile("s_wait_asynccnt 0x0" ::: "memory");
#endif
#else
    #pragma unroll
    for (int i = tid; i < 512; i += 256) {
      const int r = i >> 5;
      const int c = i & 31;
      *(float4*)(tileA + r * LDS_PITCH + c * 4) =
          *(const float4*)(gsrc + (size_t)r * IN_DIM + c * 4);
    }
#endif
  }
  __syncthreads();

  // ---- WMMA main loop: A from LDS, B from global (L2-resident, 64KB/W) ----
  v8f acc = {};
  const float* __restrict__ arow = tileA + l15 * LDS_PITCH;

  #pragma unroll 8
  for (int k0 = 0; k0 < IN_DIM; k0 += 4) {
    const int ka = k0 + half * 2;
    v2f a, b;
    a.x = arow[ka];
    a.y = arow[ka + 1];
    b.x = W[(size_t)ka * OUT_DIM + n0 + l15];
    b.y = W[(size_t)(ka + 1) * OUT_DIM + n0 + l15];
    // 8 args: (neg_a, A, neg_b, B, c_mod, C, reuse_a, reuse_b)
    acc = __builtin_amdgcn_wmma_f32_16x16x4_f32(
        false, a, false, b, (short)0, acc, false, false);
  }

  const float bb = bias[n0 + l15];
  #pragma unroll
  for (int v = 0; v < 8; ++v) {
    const int row = row0 + v + half * 8;
    out[(size_t)row * OUT_DIM + n0 + l15] = acc[v] + bb;
  }
}

// ---------------------------------------------------------------------------
// Edge phase: one wave32 per edge.
//   Each lane loads one float4 of K[src], Q[dst], V[src]  (32 lanes x 16B =
//   one full 512B row). Head = lane>>2 (4 lanes per 16-wide head). 4-lane
//   xor-shuffle reduction -> per-head score -> exp(clip(.)) -> native f32
//   atomic scatter of score*V into wV[dst] and score into z[dst].
//   Q/K/V (77 MB) are L2-resident on MI455X (192 MB L2), so this phase is
//   L2-bandwidth / atomic-throughput bound; this shape issues exactly one
//   512B burst per matrix row and 4 f32 atomics per lane.
// ---------------------------------------------------------------------------
__global__ __launch_bounds__(256) void edge_scatter(
    const int* __restrict__ src, const int* __restrict__ dst,
    const float* __restrict__ Qb, const float* __restrict__ Kb,
    const float* __restrict__ Vb,
    float* __restrict__ wV, float* __restrict__ z, int n_edges)
{
  const int wave = threadIdx.x >> 5;
  const int lane = threadIdx.x & 31;
  const int e = blockIdx.x * 8 + wave;
  if (e >= n_edges) return;

  const int s = src[e];
  const int d = dst[e];

  const float4 kv = ((const float4*)(Kb + (size_t)s * OUT_DIM))[lane];
  const float4 qv = ((const float4*)(Qb + (size_t)d * OUT_DIM))[lane];
  const float4 vv = ((const float4*)(Vb + (size_t)s * OUT_DIM))[lane];

  float part = kv.x * qv.x + kv.y * qv.y + kv.z * qv.z + kv.w * qv.w;
  // reduce across the 4 lanes of this head (wave32 shuffles)
  part += __shfl_xor(part, 1, 32);
  part += __shfl_xor(part, 2, 32);

  // score = exp(clip(dot / sqrt(16), -5, 5))
  float score = __expf(fminf(fmaxf(part * 0.25f, -5.0f), 5.0f));

  float* wrow = wV + (size_t)d * OUT_DIM + lane * 4;
  unsafeAtomicAdd(wrow + 0, vv.x * score);
  unsafeAtomicAdd(wrow + 1, vv.y * score);
  unsafeAtomicAdd(wrow + 2, vv.z * score);
  unsafeAtomicAdd(wrow + 3, vv.w * score);
  if ((lane & 3) == 0) {
    unsafeAtomicAdd(z + (size_t)d * NHEAD + (lane >> 2), score);
  }
}

// ---------------------------------------------------------------------------
// out = wV / (z + 1e-6)
// ---------------------------------------------------------------------------
__global__ void normalize_out(const float* __restrict__ wV,
                              const float* __restrict__ z,
                              float* __restrict__ out, long long total)
{
  long long i = (long long)blockIdx.x * blockDim.x + threadIdx.x;
  if (i >= total) return;
  const long long node = i >> 7;          // /128
  const int head = ((int)i & 127) >> 4;   // col/16
  out[i] = wV[i] / (z[node * NHEAD + head] + 1e-6f);
}

extern "C" void kernel_launch(void* const* d_in, const int* in_sizes, int n_in,
                              void* d_out, int out_size, void* d_ws, size_t ws_size,
                              hipStream_t stream) {
  const float* h  = (const float*)d_in[0];
  const float* Wq = (const float*)d_in[1];
  const float* bq = (const float*)d_in[2];
  const float* Wk = (const float*)d_in[3];
  const float* bk = (const float*)d_in[4];
  const float* Wv = (const float*)d_in[5];
  const float* bv = (const float*)d_in[6];
  const int* src  = (const int*)d_in[7];
  const int* dst  = (const int*)d_in[8];
  float* out = (float*)d_out;

  const int n_nodes = in_sizes[0] / IN_DIM;   // 50000
  const int n_edges = in_sizes[7];            // 1600000

  // Workspace carve (floats): Qb | Kb | Vb | wV | z   (~104 MB total)
  float* Qb = (float*)d_ws;
  float* Kb = Qb + (size_t)n_nodes * OUT_DIM;
  float* Vb = Kb + (size_t)n_nodes * OUT_DIM;
  float* wV = Vb + (size_t)n_nodes * OUT_DIM;
  float* z  = wV + (size_t)n_nodes * OUT_DIM;

  // 1) zero accumulators (wV and z are contiguous: n_nodes*(128+8) floats)
  {
    long long nz = (long long)n_nodes * (OUT_DIM + NHEAD);
    int blocks = (int)((nz + 255) / 256);
    zero_f32<<<blocks, 256, 0, stream>>>(wV, nz);
  }

  // 2) fused QKV projection via fp32 WMMA (A-tile staged by async->LDS)
  {
    dim3 grid((n_nodes + 15) / 16, 3);
    qkv_gemm_wmma<<<grid, 256, 0, stream>>>(h, Wq, bq, Wk, bk, Wv, bv,
                                            Qb, Kb, Vb, n_nodes);
  }

  // 3) edge score + scatter (one wave per edge, 8 waves per block)
  {
    int blocks = (n_edges + 7) / 8;
    edge_scatter<<<blocks, 256, 0, stream>>>(src, dst, Qb, Kb, Vb, wV, z, n_edges);
  }

  // 4) normalize into d_out
  {
    long long total = (long long)n_nodes * OUT_DIM;
    int blocks = (int)((total + 255) / 256);
    normalize_out<<<blocks, 256, 0, stream>>>(wV, z, out, total);
  }
}